// ASCGM_30090540876360
// MI455X (gfx1250) — compile-verified
//
#include <hip/hip_runtime.h>
#include <cstddef>

// ---------------------------------------------------------------------------
// Fused 1x1conv -> median-AFFR(3x3, reflect pad) -> 1x1conv for MI455X/gfx1250
//
//   d   = w1 @ x + b1                    (per-pixel GEMM, K=64)
//   d3  = AFFR3x3(d)                     (median-thresholded weighted stencil)
//   out = w2 @ [d; d3] + b2              (per-pixel GEMM, K=128)
//
// One block = one 14x14 interior tile of one batch image. The block computes
// conv1 on the full 16x16 halo (reflection folded into the x load: since conv1
// is pointwise, d(reflect(h,w)) == conv1(x(reflect(h,w)))), so no inter-block
// d traffic is needed and d never touches HBM (saves ~67 MB round trip).
// GEMMs use V_WMMA_F32_16X16X4_F32 (f32 in/out, matches reference precision).
// ---------------------------------------------------------------------------

#define C_IN   64
#define IMG_H  128
#define IMG_W  128
#define TILE   14            // interior tile edge
#define HALO   16            // TILE + 2 (stencil radius 1 on each side)
#define NPIX   256           // HALO*HALO pixels per tile
#define NTILES 10            // ceil(128/14)
#define BATCH  8

#define LDS_FLOATS (2 * C_IN * NPIX)      // x/d3 region (aliased) + d region
#define LDS_BYTES  (LDS_FLOATS * 4)       // 131072 bytes

typedef __attribute__((ext_vector_type(2))) float v2f;   // A/B frag: 16x4 f32
typedef __attribute__((ext_vector_type(8))) float v8f;   // C/D frag: 16x16 f32

__device__ __forceinline__ void mnmx(float& a, float& b) {
    float lo = fminf(a, b);
    float hi = fmaxf(a, b);
    a = lo; b = hi;
}

__device__ __forceinline__ int reflect_idx(int i, int n) {
    i = (i < 0) ? -i : i;                 // ReflectionPad2d(1): -1 -> 1
    i = (i > n - 1) ? (2 * (n - 1) - i) : i;  // n -> n-2
    return i;
}

__global__ void __launch_bounds__(256, 2)
affr_fused_kernel(const float* __restrict__ x,
                  const float* __restrict__ w1,
                  const float* __restrict__ b1,
                  const float* __restrict__ w2,
                  const float* __restrict__ b2,
                  float* __restrict__ out)
{
    extern __shared__ float lds[];
    float* ldsX  = lds;                   // 64 KB: x tile   (phase 1-2)
    float* ldsD  = lds + C_IN * NPIX;     // 64 KB: d tile   (phase 2-4)
    float* ldsD3 = lds;                   // alias of ldsX: d3 (phase 3-4)

    const int tid  = threadIdx.x;         // 256 threads = 8 wave32
    const int lane = tid & 31;
    const int wv   = tid >> 5;            // wave id 0..7
    const int lr   = lane & 15;           // M-row / N-col inside a fragment
    const int kh   = (lane >> 4) << 1;    // K offset (0|2) for A/B frags
    const int oh   = (lane >> 4) << 3;    // M offset (0|8) for C/D frags
    const int nt0  = wv * 2;              // this wave's first N-tile (= halo row)

    const int h0 = blockIdx.y * TILE;     // interior tile origin
    const int w0 = blockIdx.x * TILE;
    const int bz = blockIdx.z;            // batch index

    // ---- Phase 1: load x halo tile (reflection at image borders) into LDS --
    {
        const int ph = tid >> 4;          // halo row of this thread's pixel
        const int pw = tid & 15;          // halo col
        const int gh = reflect_idx(h0 - 1 + ph, IMG_H);
        const int gw = reflect_idx(w0 - 1 + pw, IMG_W);
        const float* xb = x + ((size_t)bz * C_IN) * (IMG_H * IMG_W)
                            + (size_t)gh * IMG_W + gw;
        #pragma unroll 8
        for (int c = 0; c < C_IN; ++c)
            ldsX[c * NPIX + tid] = xb[(size_t)c * (IMG_H * IMG_W)];
    }
    __syncthreads();

    // ---- Phase 2: GEMM1  d = w1 @ x + b1  on the full 16x16 halo -----------
    // Wave: 4 M-tiles (all 64 out-chan) x 2 N-tiles (2 halo rows). K=64.
    v8f acc[2][4];
    #pragma unroll
    for (int i = 0; i < 2; ++i)
        #pragma unroll
        for (int mt = 0; mt < 4; ++mt)
            #pragma unroll
            for (int v = 0; v < 8; ++v)
                acc[i][mt][v] = b1[mt * 16 + oh + v];

    for (int kc = 0; kc < 16; ++kc) {     // 16 K-chunks of 4
        v2f afr[4];
        #pragma unroll
        for (int mt = 0; mt < 4; ++mt)    // contiguous K pair -> b64 load
            afr[mt] = *(const v2f*)(w1 + (mt * 16 + lr) * C_IN + kc * 4 + kh);
        v2f bfr[2];
        #pragma unroll
        for (int i = 0; i < 2; ++i) {
            const int pix = (nt0 + i) * 16 + lr;
            bfr[i].x = ldsX[(kc * 4 + kh + 0) * NPIX + pix];
            bfr[i].y = ldsX[(kc * 4 + kh + 1) * NPIX + pix];
        }
        #pragma unroll
        for (int i = 0; i < 2; ++i)
            #pragma unroll
            for (int mt = 0; mt < 4; ++mt)
                acc[i][mt] = __builtin_amdgcn_wmma_f32_16x16x4_f32(
                    false, afr[mt], false, bfr[i], (short)0, acc[i][mt],
                    false, false);
    }

    #pragma unroll
    for (int i = 0; i < 2; ++i)
        #pragma unroll
        for (int mt = 0; mt < 4; ++mt)
            #pragma unroll
            for (int v = 0; v < 8; ++v)
                ldsD[(mt * 16 + oh + v) * NPIX + (nt0 + i) * 16 + lr] =
                    acc[i][mt][v];
    __syncthreads();

    // ---- Phase 3: AFFR stencil -> d3 (interior 14x14, borders zeroed) ------
    {
        const int ph = tid >> 4;
        const int pw = tid & 15;
        const bool interior = (ph >= 1) && (ph <= TILE) &&
                              (pw >= 1) && (pw <= TILE);
        for (int c = 0; c < C_IN; ++c) {
            float res = 0.0f;
            if (interior) {
                const float* dc = ldsD + c * NPIX + ph * HALO + pw;
                const float ctr = dc[0];
                float f0 = dc[-HALO - 1] - ctr, f1 = dc[-HALO] - ctr,
                      f2 = dc[-HALO + 1] - ctr;
                float f3 = dc[-1] - ctr,        f5 = dc[1] - ctr;
                float f6 = dc[HALO - 1] - ctr,  f7 = dc[HALO] - ctr,
                      f8 = dc[HALO + 1] - ctr;
                float a0 = fabsf(f0), a1 = fabsf(f1), a2 = fabsf(f2);
                float a3 = fabsf(f3), a5 = fabsf(f5);
                float a6 = fabsf(f6), a7 = fabsf(f7), a8 = fabsf(f8);
                // lower median of the 9 abs-diffs (center contributes 0),
                // 19-exchange Paeth min/max network
                float m0 = a0, m1 = a1, m2 = a2, m3 = a3, m4 = 0.0f,
                      m5 = a5, m6 = a6, m7 = a7, m8 = a8;
                mnmx(m1, m2); mnmx(m4, m5); mnmx(m7, m8);
                mnmx(m0, m1); mnmx(m3, m4); mnmx(m6, m7);
                mnmx(m1, m2); mnmx(m4, m5); mnmx(m7, m8);
                mnmx(m0, m3); mnmx(m5, m8); mnmx(m4, m7);
                mnmx(m3, m6); mnmx(m1, m4); mnmx(m2, m5);
                mnmx(m4, m7); mnmx(m4, m2); mnmx(m6, m4);
                mnmx(m4, m2);
                const float med = m4;
                // max of kept |diff| == med, so s_k = a_k / med
                const float inv = 1.0f / med;   // med==0 -> NaN, as reference
                res  = (a0 <= med) ? f0 * (1.0f - a0 * inv) : 0.0f;
                res += (a1 <= med) ? f1 * (1.0f - a1 * inv) : 0.0f;
                res += (a2 <= med) ? f2 * (1.0f - a2 * inv) : 0.0f;
                res += (a3 <= med) ? f3 * (1.0f - a3 * inv) : 0.0f;
                res += (a5 <= med) ? f5 * (1.0f - a5 * inv) : 0.0f;
                res += (a6 <= med) ? f6 * (1.0f - a6 * inv) : 0.0f;
                res += (a7 <= med) ? f7 * (1.0f - a7 * inv) : 0.0f;
                res += (a8 <= med) ? f8 * (1.0f - a8 * inv) : 0.0f;
            }
            ldsD3[c * NPIX + tid] = res;      // borders written as 0
        }
    }
    __syncthreads();

    // ---- Phase 4: GEMM2  out = w2[:, :64]@d + w2[:, 64:]@d3 + b2 -----------
    v8f acc2[2][4];
    #pragma unroll
    for (int i = 0; i < 2; ++i)
        #pragma unroll
        for (int mt = 0; mt < 4; ++mt)
            #pragma unroll
            for (int v = 0; v < 8; ++v)
                acc2[i][mt][v] = b2[mt * 16 + oh + v];

    for (int half = 0; half < 2; ++half) {
        const float* bsrc = half ? ldsD3 : ldsD;
        for (int kc = 0; kc < 16; ++kc) {
            v2f afr[4];
            #pragma unroll
            for (int mt = 0; mt < 4; ++mt)
                afr[mt] = *(const v2f*)(w2 + (mt * 16 + lr) * (2 * C_IN)
                                           + half * C_IN + kc * 4 + kh);
            v2f bfr[2];
            #pragma unroll
            for (int i = 0; i < 2; ++i) {
                const int pix = (nt0 + i) * 16 + lr;
                bfr[i].x = bsrc[(kc * 4 + kh + 0) * NPIX + pix];
                bfr[i].y = bsrc[(kc * 4 + kh + 1) * NPIX + pix];
            }
            #pragma unroll
            for (int i = 0; i < 2; ++i)
                #pragma unroll
                for (int mt = 0; mt < 4; ++mt)
                    acc2[i][mt] = __builtin_amdgcn_wmma_f32_16x16x4_f32(
                        false, afr[mt], false, bfr[i], (short)0, acc2[i][mt],
                        false, false);
        }
    }

    // ---- Stores: interior pixels only (each output written by one tile) ----
    #pragma unroll
    for (int i = 0; i < 2; ++i) {
        const int nt = nt0 + i;               // halo row == N-tile index
        const int gh = h0 + nt - 1;
        const int gw = w0 + lr - 1;
        const bool rowOK = (nt >= 1) && (nt <= TILE) && (gh < IMG_H);
        const bool colOK = (lr >= 1) && (lr <= TILE) && (gw < IMG_W);
        if (rowOK && colOK) {
            float* ob = out + (((size_t)bz * C_IN) * IMG_H + gh) * IMG_W + gw;
            #pragma unroll
            for (int mt = 0; mt < 4; ++mt)
                #pragma unroll
                for (int v = 0; v < 8; ++v)
                    ob[(size_t)(mt * 16 + oh + v) * (IMG_H * IMG_W)] =
                        acc2[i][mt][v];
        }
    }
}

extern "C" void kernel_launch(void* const* d_in, const int* in_sizes, int n_in,
                              void* d_out, int out_size, void* d_ws, size_t ws_size,
                              hipStream_t stream) {
    (void)in_sizes; (void)n_in; (void)d_ws; (void)ws_size; (void)out_size;
    const float* x  = (const float*)d_in[0];   // (8, 64, 128, 128)
    const float* w1 = (const float*)d_in[1];   // (64, 64)
    const float* b1 = (const float*)d_in[2];   // (64,)
    const float* w2 = (const float*)d_in[3];   // (64, 128)
    const float* b2 = (const float*)d_in[4];   // (64,)
    float* out = (float*)d_out;                // (8, 64, 128, 128)

    // 128 KB dynamic LDS (> 64 KB default cap) — idempotent, host-side only,
    // safe under graph capture.
    (void)hipFuncSetAttribute((const void*)affr_fused_kernel,
                              hipFuncAttributeMaxDynamicSharedMemorySize,
                              (int)LDS_BYTES);

    dim3 grid(NTILES, NTILES, BATCH);          // 10 x 10 x 8 = 800 blocks
    affr_fused_kernel<<<grid, 256, LDS_BYTES, stream>>>(x, w1, b1, w2, b2, out);
}